// PsRoIPool2D_30666066493718
// MI455X (gfx1250) — compile-verified
//
#include <hip/hip_runtime.h>

typedef __attribute__((ext_vector_type(2))) float v2f;
typedef __attribute__((ext_vector_type(8))) float v8f;

#define KPH 7
#define KPW 7
#define KG 7
#define KD 8
#define KC 392          // D*G*G
#define KH 96
#define KW 96
#define KR 512
#define KSCALE 0.0625f
#define NOUT 56         // D*PW outputs per (roi, p) row

// One block (4 waves, 128 threads) per (roi r, bin-row p).
// Wave g reduces outputs o = g*16 + m, m = 0..15 (o < 56 valid).
// Each output's bin rectangle is split into 4 interleaved k-slots; lane half
// (lane>>4) owns k = 2*half and 2*half+1 (the 16x16x4 f32 A-matrix layout).
// One V_WMMA_F32_16X16X4_F32 against an all-ones B reduces the 4 slots.
__global__ __launch_bounds__(128) void psroi_wmma_kernel(
    const float* __restrict__ feat,   // [B, C, H, W]
    const float* __restrict__ rois,   // [R, 5]
    float* __restrict__ out)          // [R, D, PH, PW]
{
    __shared__ float red[4][16];

    const int tid  = threadIdx.x;
    const int wave = tid >> 5;      // WMMA group g (0..3)
    const int lane = tid & 31;
    const int half = lane >> 4;
    const int m    = lane & 15;

    const int blk = blockIdx.x;     // r*PH + p
    const int r = blk / KPH;
    const int p = blk - r * KPH;

    // --- ROI decode (uniform per block; jnp.round == RNE == rintf) ---
    const float* roi = rois + r * 5;
    const int   b  = (int)roi[0];
    const float x1 = rintf(roi[1]) * KSCALE;
    const float y1 = rintf(roi[2]) * KSCALE;
    const float x2 = rintf(roi[3] + 1.0f) * KSCALE;
    const float y2 = rintf(roi[4] + 1.0f) * KSCALE;
    const float rw = fmaxf(x2 - x1, 0.1f);
    const float rh = fmaxf(y2 - y1, 0.1f);
    const float binh = rh * (1.0f / KPH);
    const float binw = rw * (1.0f / KPW);

    // Row-bin extents (uniform per block)
    const int sh = (int)fminf(fmaxf(floorf((float)p * binh + y1), 0.0f), (float)KH);
    const int eh = (int)fminf(fmaxf(ceilf((float)(p + 1) * binh + y1), 0.0f), (float)KH);
    int nh = eh - sh; if (nh < 0) nh = 0;

    // --- Gather phase: accumulate this lane's two k-slots ---
    const int o = wave * 16 + m;    // (d,q) flat output index for this row
    float a0 = 0.0f, a1 = 0.0f;
    if (o < NOUT) {
        const int d = o / KPW;
        const int q = o - d * KPW;
        const int sw = (int)fminf(fmaxf(floorf((float)q * binw + x1), 0.0f), (float)KW);
        const int ew = (int)fminf(fmaxf(ceilf((float)(q + 1) * binw + x1), 0.0f), (float)KW);
        int nw = ew - sw; if (nw < 0) nw = 0;

        const int c = d * (KG * KG) + p * KG + q;
        const float* base = feat + (size_t)(b * KC + c) * (KH * KW);
        const int area = nh * nw;
        for (int j = 2 * half; j < area; j += 4) {
            int dh = j / nw;
            int dw = j - dh * nw;
            a0 += base[(sh + dh) * KW + (sw + dw)];
            int j1 = j + 1;
            if (j1 < area) {
                int dh1 = j1 / nw;
                int dw1 = j1 - dh1 * nw;
                a1 += base[(sh + dh1) * KW + (sw + dw1)];
            }
        }
    }

    // --- WMMA 4-way cross-lane reduction: D[m,n] = sum_k A[m,k] ---
    v2f A; A[0] = a0; A[1] = a1;
    v2f Bones; Bones[0] = 1.0f; Bones[1] = 1.0f;   // all-ones B, layout-agnostic
    v8f acc = {};
    acc = __builtin_amdgcn_wmma_f32_16x16x4_f32(
        /*neg_a=*/false, A, /*neg_b=*/false, Bones,
        /*c_mod=*/(short)0, acc, /*reuse_a=*/false, /*reuse_b=*/false);

    // D layout: VGPR t, lanes 0-15 -> M=t (N=lane), lanes 16-31 -> M=t+8.
    // Lane 0 holds rows 0..7 at N=0; lane 16 holds rows 8..15 at N=0.
    if ((lane & 15) == 0) {
#pragma unroll
        for (int t = 0; t < 8; ++t) red[wave][8 * half + t] = acc[t];
    }
    __syncthreads();

    // --- Scale + store: lanes 0..15 each own one output row ---
    if (lane < 16) {
        const int o2 = wave * 16 + lane;
        if (o2 < NOUT) {
            const int d = o2 / KPW;
            const int q = o2 - d * KPW;
            const int sw = (int)fminf(fmaxf(floorf((float)q * binw + x1), 0.0f), (float)KW);
            const int ew = (int)fminf(fmaxf(ceilf((float)(q + 1) * binw + x1), 0.0f), (float)KW);
            int nw = ew - sw; if (nw < 0) nw = 0;
            const float inv = 1.0f / ((float)(nh > 1 ? nh : 1) * (float)(nw > 1 ? nw : 1));
            out[(((size_t)r * KD + d) * KPH + p) * KPW + q] = red[wave][lane] * inv;
        }
    }
}

extern "C" void kernel_launch(void* const* d_in, const int* in_sizes, int n_in,
                              void* d_out, int out_size, void* d_ws, size_t ws_size,
                              hipStream_t stream) {
    const float* feat = (const float*)d_in[0];   // [4, 392, 96, 96] f32
    const float* rois = (const float*)d_in[1];   // [512, 5] f32
    float* out = (float*)d_out;                  // [512, 8, 7, 7] f32

    const int blocks = KR * KPH;                 // 3584 blocks, one per (r, p)
    psroi_wmma_kernel<<<blocks, 128, 0, stream>>>(feat, rois, out);
    (void)in_sizes; (void)n_in; (void)out_size; (void)d_ws; (void)ws_size;
}